// ConvolutionalVectorQuantizedVAE_28698971472439
// MI455X (gfx1250) — compile-verified
//
#include <hip/hip_runtime.h>
#include <hip/hip_bf16.h>
#include <math.h>

// ---------------------------------------------------------------------------
// VQ-VAE forward for MI455X (gfx1250, wave32).
// Convs: fp32 direct (memory-bound; tiny channel counts) with fused BN-stat
//        accumulation (training-mode batch stats) + global_prefetch.
// VQ:    V_WMMA_F32_16X16X4_F32 (exact fp32 on the matrix unit) with the
//        codebook staged into LDS via GLOBAL_LOAD_ASYNC_TO_LDS_B128.
// ---------------------------------------------------------------------------

typedef __attribute__((ext_vector_type(2))) float v2f;
typedef __attribute__((ext_vector_type(8))) float v8f;
typedef int v4i __attribute__((vector_size(16)));
typedef __attribute__((address_space(1))) v4i gv4i_t;   // global b128 unit
typedef __attribute__((address_space(3))) v4i lv4i_t;   // LDS b128 unit

#define EPS_BN 1e-5f

#if __has_builtin(__builtin_amdgcn_wmma_f32_16x16x4_f32)
#define HAVE_WMMA_F32X4 1
#else
#define HAVE_WMMA_F32X4 0
#endif

#if __has_builtin(__builtin_amdgcn_global_load_async_to_lds_b128)
#define HAVE_ASYNC_LDS 1
#else
#define HAVE_ASYNC_LDS 0
#endif

__device__ __forceinline__ void wait_async_zero() {
#if __has_builtin(__builtin_amdgcn_s_wait_asynccnt)
    __builtin_amdgcn_s_wait_asynccnt(0);
#else
    asm volatile("s_wait_asynccnt 0" ::: "memory");
#endif
}

// ------------------------------ init -----------------------------------
__global__ void init_ws_kernel(float* ws) {
    if (threadIdx.x < 512) ws[threadIdx.x] = 0.0f;
}

// --------------------- conv2d (+ BN stat accumulation) ------------------
// out[b][co][oy][ox] = sum_{ci,ky,kx} in[b][ci][iy][ix] * w[co][ci][ky][kx]
__global__ void conv2d_pre_kernel(const float* __restrict__ in, const float* __restrict__ w,
                                  float* __restrict__ out, float* __restrict__ acc,
                                  int Bn, int Cin, int H, int W, int Cout, int OH, int OW,
                                  int KH, int KW, int stride, int pad) {
    __shared__ float sw[2304];
    __shared__ float s_sum[16];
    __shared__ float s_sq[16];
    int tid = threadIdx.x;
    int wn = Cout * Cin * KH * KW;
    for (int i = tid; i < wn; i += blockDim.x) sw[i] = w[i];
    if (tid < Cout) { s_sum[tid] = 0.0f; s_sq[tid] = 0.0f; }
    __syncthreads();

    int idx = blockIdx.x * blockDim.x + tid;
    int total = Bn * Cout * OH * OW;
    int co = 0;
    float v = 0.0f;
    bool valid = idx < total;
    if (valid) {
        int ox = idx % OW; int t = idx / OW;
        int oy = t % OH;   t /= OH;
        co = t % Cout;     int b = t / Cout;
        const float* inb = in + (b * Cin) * (H * W);
        {   // prefetch this thread's first in-range input row (global_prefetch_b8)
            int iy0 = oy * stride - pad; if (iy0 < 0) iy0 = 0;
            int ix0 = ox * stride - pad; if (ix0 < 0) ix0 = 0;
            __builtin_prefetch(inb + iy0 * W + ix0, 0, 0);
        }
        const float* wc = sw + co * Cin * KH * KW;
        for (int ci = 0; ci < Cin; ++ci) {
            const float* inc = inb + ci * (H * W);
            const float* wci = wc + ci * (KH * KW);
            for (int ky = 0; ky < KH; ++ky) {
                int iy = oy * stride - pad + ky;
                if (iy < 0 || iy >= H) continue;
                const float* row = inc + iy * W;
                for (int kx = 0; kx < KW; ++kx) {
                    int ix = ox * stride - pad + kx;
                    if (ix < 0 || ix >= W) continue;
                    v = fmaf(row[ix], wci[ky * KW + kx], v);
                }
            }
        }
        out[idx] = v;
        atomicAdd(&s_sum[co], v);
        atomicAdd(&s_sq[co], v * v);
    }
    __syncthreads();
    if (tid < Cout) {
        atomicAdd(&acc[2 * tid],     s_sum[tid]);
        atomicAdd(&acc[2 * tid + 1], s_sq[tid]);
    }
}

// ---------------- conv-transpose (+ BN stat accumulation) ---------------
// w layout (torch ConvTranspose2d): (Cin, Cout, KH, KW). Gather form:
// out[b][co][oy][ox] = sum over (ci,ky,kx) with iy=(oy+pad-ky)/s when exact.
__global__ void convT2d_pre_kernel(const float* __restrict__ in, const float* __restrict__ w,
                                   float* __restrict__ out, float* __restrict__ acc,
                                   int Bn, int Cin, int H, int W, int Cout, int OH, int OW,
                                   int KH, int KW, int stride, int pad) {
    __shared__ float sw[1200];
    __shared__ float s_sum[16];
    __shared__ float s_sq[16];
    int tid = threadIdx.x;
    int wn = Cin * Cout * KH * KW;
    for (int i = tid; i < wn; i += blockDim.x) sw[i] = w[i];
    if (tid < Cout) { s_sum[tid] = 0.0f; s_sq[tid] = 0.0f; }
    __syncthreads();

    int idx = blockIdx.x * blockDim.x + tid;
    int total = Bn * Cout * OH * OW;
    int co = 0;
    float v = 0.0f;
    bool valid = idx < total;
    if (valid) {
        int ox = idx % OW; int t = idx / OW;
        int oy = t % OH;   t /= OH;
        co = t % Cout;     int b = t / Cout;
        const float* inb = in + (b * Cin) * (H * W);
        {
            int iy0 = (oy + pad) / stride; if (iy0 >= H) iy0 = H - 1;
            int ix0 = (ox + pad) / stride; if (ix0 >= W) ix0 = W - 1;
            __builtin_prefetch(inb + iy0 * W + ix0, 0, 0);
        }
        for (int ky = 0; ky < KH; ++ky) {
            int ny = oy + pad - ky;
            if (ny < 0 || (ny % stride) != 0) continue;
            int iy = ny / stride;
            if (iy >= H) continue;
            for (int kx = 0; kx < KW; ++kx) {
                int nx = ox + pad - kx;
                if (nx < 0 || (nx % stride) != 0) continue;
                int ix = nx / stride;
                if (ix >= W) continue;
                for (int ci = 0; ci < Cin; ++ci) {
                    float xin = inb[ci * (H * W) + iy * W + ix];
                    float wv  = sw[((ci * Cout + co) * KH + ky) * KW + kx];
                    v = fmaf(xin, wv, v);
                }
            }
        }
        out[idx] = v;
        atomicAdd(&s_sum[co], v);
        atomicAdd(&s_sq[co], v * v);
    }
    __syncthreads();
    if (tid < Cout) {
        atomicAdd(&acc[2 * tid],     s_sum[tid]);
        atomicAdd(&acc[2 * tid + 1], s_sq[tid]);
    }
}

// -------------------------- BN (batch stats) + ReLU ---------------------
__global__ void bn_relu_kernel(const float* __restrict__ raw, const float* __restrict__ acc,
                               const float* __restrict__ g, const float* __restrict__ bb,
                               float* __restrict__ out, float* __restrict__ z_out,
                               int Bn, int C, int HW, float invN) {
    int idx = blockIdx.x * blockDim.x + threadIdx.x;
    int total = Bn * C * HW;
    if (idx >= total) return;
    int pix = idx % HW; int t = idx / HW;
    int c = t % C;      int b = t / C;
    float m   = acc[2 * c] * invN;
    float var = acc[2 * c + 1] * invN - m * m;     // biased variance (jnp.var)
    float y = g[c] * (raw[idx] - m) * rsqrtf(var + EPS_BN) + bb[c];
    y = fmaxf(y, 0.0f);
    out[idx] = y;
    if (z_out) z_out[(b * HW + pix) * 4 + c] = y;  // token-major [N,4] copy
}

// ------------------------------ VQ via WMMA -----------------------------
// One wave handles 16 tokens vs all 512 codes: 32 x V_WMMA_F32_16X16X4_F32.
// A (16x4): lane l<16 -> {z[t][0],z[t][1]}, lane l>=16 -> {z[t][2],z[t][3]}.
// B (4x16): lane l<16 -> {e[n][0],e[n][1]}, lane l>=16 -> {e[n][2],e[n][3]}.
// D (16x16): VGPR r, lane l -> row r+8*(l>>4), col l&15.
// argmin_n(||z||^2 - 2 z.e + ||e||^2) == argmin_n(||e||^2 - 2 z.e).
// Codebook (8KB) staged to LDS via async loads; ||e||^2 precomputed in LDS.
__global__ void vq_wmma_kernel(const float* __restrict__ z, const float* __restrict__ emb,
                               float* __restrict__ q_out, float* __restrict__ loss_acc) {
    __shared__ __align__(16) float s_emb[2048];   // 512 codes x 4 dims
    __shared__ float s_ee[512];
    int tid = threadIdx.x;

#if HAVE_ASYNC_LDS
    {
        // 512 x b128 transfers; 256 threads issue 2 each (ASYNCcnt-tracked)
        __builtin_amdgcn_global_load_async_to_lds_b128(
            (gv4i_t*)(emb + tid * 4), (lv4i_t*)(s_emb + tid * 4), 0, 0);
        __builtin_amdgcn_global_load_async_to_lds_b128(
            (gv4i_t*)(emb + 1024 + tid * 4), (lv4i_t*)(s_emb + 1024 + tid * 4), 0, 0);
        wait_async_zero();
    }
#else
    for (int i = tid; i < 2048; i += 256) s_emb[i] = emb[i];
#endif
    __syncthreads();
    for (int i = tid; i < 512; i += 256) {
        const float* er = s_emb + i * 4;
        s_ee[i] = er[0]*er[0] + er[1]*er[1] + er[2]*er[2] + er[3]*er[3];
    }
    __syncthreads();

    int lane = tid & 31;
    int wave = blockIdx.x * (blockDim.x >> 5) + (tid >> 5);
    int tok_base = wave * 16;
    int half = lane >> 4;   // 0: K=0..1 / rows 0..7 of D ; 1: K=2..3 / rows 8..15
    int li   = lane & 15;

    float best[8];
    int   bidx[8];
#pragma unroll
    for (int r = 0; r < 8; ++r) { best[r] = 3.4e38f; bidx[r] = 0; }

#if HAVE_WMMA_F32X4
    int t = tok_base + li;
    v2f a;
    a.x = z[t * 4 + half * 2 + 0];
    a.y = z[t * 4 + half * 2 + 1];
    for (int n0 = 0; n0 < 512; n0 += 16) {
        int n = n0 + li;
        v2f bv;
        bv.x = s_emb[n * 4 + half * 2 + 0];
        bv.y = s_emb[n * 4 + half * 2 + 1];
        float ee = s_ee[n];
        v8f p = {0.f, 0.f, 0.f, 0.f, 0.f, 0.f, 0.f, 0.f};
        p = __builtin_amdgcn_wmma_f32_16x16x4_f32(false, a, false, bv, (short)0, p,
                                                  false, false);
#pragma unroll
        for (int r = 0; r < 8; ++r) {
            float s = ee - 2.0f * p[r];
            if (s < best[r]) { best[r] = s; bidx[r] = n; }
        }
    }
#else
    // Scalar fallback (same lane<->(row,col) mapping, no matrix unit).
    for (int n0 = 0; n0 < 512; n0 += 16) {
        int n = n0 + li;
        const float* er = s_emb + n * 4;
        float ee = s_ee[n];
#pragma unroll
        for (int r = 0; r < 8; ++r) {
            int row = half * 8 + r;
            const float* zr = z + (tok_base + row) * 4;
            float dot = zr[0]*er[0] + zr[1]*er[1] + zr[2]*er[2] + zr[3]*er[3];
            float s = ee - 2.0f * dot;
            if (s < best[r]) { best[r] = s; bidx[r] = n; }
        }
    }
#endif

    // 16-lane butterfly min+argmin (masks 1..8 stay inside each half-wave).
#pragma unroll
    for (int m = 1; m < 16; m <<= 1) {
#pragma unroll
        for (int r = 0; r < 8; ++r) {
            float ov = __shfl_xor(best[r], m, 32);
            int   oi = __shfl_xor(bidx[r], m, 32);
            if (ov < best[r] || (ov == best[r] && oi < bidx[r])) {
                best[r] = ov; bidx[r] = oi;
            }
        }
    }

    // Writeback: lanes 0..7 handle rows 0..7, lanes 16..23 handle rows 8..15.
    if (li < 8) {
        int row = half * 8 + li;
        int tok = tok_base + row;
        int c = bidx[li];
        int b = tok >> 10;
        int rem = tok & 1023;
        float lsum = 0.0f;
#pragma unroll
        for (int d = 0; d < 4; ++d) {
            float e  = s_emb[c * 4 + d];
            float zv = z[tok * 4 + d];
            q_out[(b * 4 + d) * 1024 + rem] = e;     // NCHW quantized
            float diff = e - zv;
            lsum += diff * diff;
        }
        atomicAdd(loss_acc, lsum);
    }
}

// ------------------- final convT + bias + sigmoid -----------------------
__global__ void convT_sigmoid_kernel(const float* __restrict__ in, const float* __restrict__ w,
                                     const float* __restrict__ bias, float* __restrict__ out,
                                     int Bn, int Cin, int H, int W, int Cout, int OH, int OW,
                                     int KH, int KW, int stride, int pad) {
    __shared__ float sw[128];
    int tid = threadIdx.x;
    int wn = Cin * Cout * KH * KW;
    for (int i = tid; i < wn; i += blockDim.x) sw[i] = w[i];
    __syncthreads();

    int idx = blockIdx.x * blockDim.x + tid;
    int total = Bn * Cout * OH * OW;
    if (idx >= total) return;
    int ox = idx % OW; int t = idx / OW;
    int oy = t % OH;   t /= OH;
    int co = t % Cout; int b = t / Cout;
    const float* inb = in + (b * Cin) * (H * W);
    float v = bias[co];
    for (int ky = 0; ky < KH; ++ky) {
        int ny = oy + pad - ky;
        if (ny < 0 || (ny % stride) != 0) continue;
        int iy = ny / stride;
        if (iy >= H) continue;
        for (int kx = 0; kx < KW; ++kx) {
            int nx = ox + pad - kx;
            if (nx < 0 || (nx % stride) != 0) continue;
            int ix = nx / stride;
            if (ix >= W) continue;
            for (int ci = 0; ci < Cin; ++ci) {
                v = fmaf(inb[ci * (H * W) + iy * W + ix],
                         sw[((ci * Cout + co) * KH + ky) * KW + kx], v);
            }
        }
    }
    out[idx] = 1.0f / (1.0f + expf(-v));
}

// --------------------------- loss finalize ------------------------------
__global__ void finalize_losses_kernel(const float* loss_acc, float* out_losses) {
    // codebook_loss == commitment_loss numerically (stop_gradient affects grads only)
    float v = loss_acc[0] * (1.0f / 1048576.0f);   // sum over 65536 tokens * 4 dims, scaled by 1/1048576
    out_losses[0] = v;
    out_losses[1] = v;
}

// ---------------------------------------------------------------------------
extern "C" void kernel_launch(void* const* d_in, const int* in_sizes, int n_in,
                              void* d_out, int out_size, void* d_ws, size_t ws_size,
                              hipStream_t stream) {
    const float* x        = (const float*)d_in[0];
    const float* enc_w1   = (const float*)d_in[1];
    const float* enc_g1   = (const float*)d_in[2];
    const float* enc_b1   = (const float*)d_in[3];
    const float* enc_w2   = (const float*)d_in[4];
    const float* enc_g2   = (const float*)d_in[5];
    const float* enc_b2   = (const float*)d_in[6];
    const float* enc_w3   = (const float*)d_in[7];
    const float* enc_g3   = (const float*)d_in[8];
    const float* enc_b3   = (const float*)d_in[9];
    const float* emb      = (const float*)d_in[10];
    const float* dec_w1   = (const float*)d_in[11];
    const float* dec_g1   = (const float*)d_in[12];
    const float* dec_b1   = (const float*)d_in[13];
    const float* dec_w2   = (const float*)d_in[14];
    const float* dec_g2   = (const float*)d_in[15];
    const float* dec_b2   = (const float*)d_in[16];
    const float* dec_w3   = (const float*)d_in[17];
    const float* dec_bias3= (const float*)d_in[18];

    float* out = (float*)d_out;
    float* ws  = (float*)d_ws;

    // d_out layout (floats): latents | quantized | decoded | 2 losses
    const int OUT_LAT  = 0;
    const int OUT_Q    = 262144;
    const int OUT_DEC  = 524288;
    const int OUT_LOSS = 524288 + 12582912;

    // workspace layout (floats)
    const int OFF_ACC  = 0;        // 5 layers * 32 floats (sum, sumsq per ch)
    const int OFF_LOSS = 256;
    const int OFF_Z    = 1024;                 // 262144 floats, [N,4]
    const int SZ_BIG   = 8520192;              // 64*8*129*129 (largest raw/act)
    const int OFF_BUF0 = 263168;
    const int OFF_BUF1 = OFF_BUF0 + SZ_BIG;
    const int OFF_ACT0 = OFF_BUF1 + SZ_BIG;
    const int OFF_ACT1 = OFF_ACT0 + SZ_BIG;    // 4326400 floats needed

    float* acc  = ws + OFF_ACC;
    float* loss = ws + OFF_LOSS;
    float* zbuf = ws + OFF_Z;
    float* BUF0 = ws + OFF_BUF0;
    float* BUF1 = ws + OFF_BUF1;
    float* ACT0 = ws + OFF_ACT0;
    float* ACT1 = ws + OFF_ACT1;

    const int TB = 256;
    auto nb = [](int total) { return (total + 255) / 256; };

    // zero stat accumulators + loss
    init_ws_kernel<<<1, 512, 0, stream>>>(ws);

    // ---------------- encoder ----------------
    // L1: [64,3,256,256] -> [64,8,127,127], k5 s2 p1
    {
        int total = 64 * 8 * 127 * 127;
        conv2d_pre_kernel<<<nb(total), TB, 0, stream>>>(x, enc_w1, BUF0, acc + 0,
            64, 3, 256, 256, 8, 127, 127, 5, 5, 2, 1);
        bn_relu_kernel<<<nb(total), TB, 0, stream>>>(BUF0, acc + 0, enc_g1, enc_b1,
            ACT0, nullptr, 64, 8, 127 * 127, 1.0f / (64.0f * 127.0f * 127.0f));
    }
    // L2: [64,8,127,127] -> [64,16,64,64], k3 s2 p1
    {
        int total = 64 * 16 * 64 * 64;
        conv2d_pre_kernel<<<nb(total), TB, 0, stream>>>(ACT0, enc_w2, BUF1, acc + 32,
            64, 8, 127, 127, 16, 64, 64, 3, 3, 2, 1);
        bn_relu_kernel<<<nb(total), TB, 0, stream>>>(BUF1, acc + 32, enc_g2, enc_b2,
            ACT1, nullptr, 64, 16, 64 * 64, 1.0f / (64.0f * 64.0f * 64.0f));
    }
    // L3: [64,16,64,64] -> [64,4,32,32], k3 s2 p1 ; writes latents + z copy
    {
        int total = 64 * 4 * 32 * 32;
        conv2d_pre_kernel<<<nb(total), TB, 0, stream>>>(ACT1, enc_w3, BUF0, acc + 64,
            64, 16, 64, 64, 4, 32, 32, 3, 3, 2, 1);
        bn_relu_kernel<<<nb(total), TB, 0, stream>>>(BUF0, acc + 64, enc_g3, enc_b3,
            out + OUT_LAT, zbuf, 64, 4, 32 * 32, 1.0f / (64.0f * 32.0f * 32.0f));
    }

    // ---------------- VQ (WMMA f32 16x16x4) ----------------
    // 65536 tokens / 16 per wave = 4096 waves = 512 blocks of 8 waves
    vq_wmma_kernel<<<512, 256, 0, stream>>>(zbuf, emb, out + OUT_Q, loss);

    // ---------------- decoder ----------------
    // T1: [64,4,32,32] -> [64,16,65,65], k3 s2 p0
    {
        int total = 64 * 16 * 65 * 65;
        convT2d_pre_kernel<<<nb(total), TB, 0, stream>>>(out + OUT_Q, dec_w1, BUF0, acc + 96,
            64, 4, 32, 32, 16, 65, 65, 3, 3, 2, 0);
        bn_relu_kernel<<<nb(total), TB, 0, stream>>>(BUF0, acc + 96, dec_g1, dec_b1,
            ACT1, nullptr, 64, 16, 65 * 65, 1.0f / (64.0f * 65.0f * 65.0f));
    }
    // T2: [64,16,65,65] -> [64,8,129,129], k3 s2 p1
    {
        int total = 64 * 8 * 129 * 129;
        convT2d_pre_kernel<<<nb(total), TB, 0, stream>>>(ACT1, dec_w2, BUF1, acc + 128,
            64, 16, 65, 65, 8, 129, 129, 3, 3, 2, 1);
        bn_relu_kernel<<<nb(total), TB, 0, stream>>>(BUF1, acc + 128, dec_g2, dec_b2,
            ACT0, nullptr, 64, 8, 129 * 129, 1.0f / (64.0f * 129.0f * 129.0f));
    }
    // T3: [64,8,129,129] -> [64,3,256,256], k2 s2 p1, + bias + sigmoid
    {
        int total = 64 * 3 * 256 * 256;
        convT_sigmoid_kernel<<<nb(total), TB, 0, stream>>>(ACT0, dec_w3, dec_bias3,
            out + OUT_DEC, 64, 8, 129, 129, 3, 256, 256, 2, 2, 2, 1);
    }

    finalize_losses_kernel<<<1, 1, 0, stream>>>(loss, out + OUT_LOSS);
}